// VariationalGCNDecoder_6030134083625
// MI455X (gfx1250) — compile-verified
//
#include <hip/hip_runtime.h>
#include <hip/hip_bf16.h>

typedef __attribute__((ext_vector_type(2))) float v2f;
typedef __attribute__((ext_vector_type(8))) float v8f;

#define NN 262144      // nodes
#define BATCH 256

__device__ __forceinline__ float lrelu(float x) { return x >= 0.f ? x : 0.01f * x; }

// ---------------- degree / normalization ----------------
__global__ void k_fill1(float* p, int n) {
    int i = blockIdx.x * blockDim.x + threadIdx.x;
    if (i < n) p[i] = 1.0f;                       // self-loop contributes 1 to degree
}

__global__ void k_deg(const long long* __restrict__ col, float* deg, int e) {
    int i = blockIdx.x * blockDim.x + threadIdx.x;
    if (i < e) atomicAdd(&deg[col[i]], 1.0f);
}

__global__ void k_rsqrt(float* p, int n) {
    int i = blockIdx.x * blockDim.x + threadIdx.x;
    if (i < n) p[i] = rsqrtf(p[i]);               // deg >= 1 always (self loop)
}

// ---------------- dense node transform via V_WMMA_F32_16X16X4_F32 ----------------
// out[16 nodes, OUTF] = x[16 nodes, INF] @ W[INF, OUTF]  (K chained in steps of 4)
template <int INF, int OUTF, bool BIAS_ACT>
__global__ void k_transform(const float* __restrict__ x, const float* __restrict__ W,
                            const float* __restrict__ bias, float* __restrict__ out,
                            int nTiles) {
    int wave = (blockIdx.x * blockDim.x + threadIdx.x) >> 5;
    if (wave >= nTiles) return;                   // wave-uniform: EXEC stays all-1s
    int lane  = threadIdx.x & 31;
    int mrow  = lane & 15;
    int hi    = lane >> 4;        // lane half
    int khalf = hi * 2;
    int n     = lane & 15;
    int base  = wave * 16;

    v8f c = {};
    constexpr int KPAD = (INF + 3) & ~3;
#pragma unroll
    for (int k0 = 0; k0 < KPAD; k0 += 4) {
        int kA = k0 + khalf;      // A: VGPR v, lane-half h holds K = k0 + 2h + v
        v2f a, b;
        a.x = (kA + 0 < INF) ? x[(base + mrow) * INF + kA + 0] : 0.f;
        a.y = (kA + 1 < INF) ? x[(base + mrow) * INF + kA + 1] : 0.f;
        b.x = (kA + 0 < INF && n < OUTF) ? W[(kA + 0) * OUTF + n] : 0.f;
        b.y = (kA + 1 < INF && n < OUTF) ? W[(kA + 1) * OUTF + n] : 0.f;
        c = __builtin_amdgcn_wmma_f32_16x16x4_f32(false, a, false, b, (short)0, c,
                                                  false, false);
    }
    if (n < OUTF) {
#pragma unroll
        for (int r = 0; r < 8; ++r) {
            int m = r + hi * 8;   // C/D: VGPR r -> rows r (lanes 0-15) / r+8 (16-31)
            float v = c[r];
            if (BIAS_ACT) v = lrelu(v + bias[n]);
            out[(base + m) * OUTF + n] = v;
        }
    }
}

// ---------------- accumulator init: bias + self-loop message ----------------
template <int OUTF>
__global__ void k_acc_init(const float* __restrict__ ht, const float* __restrict__ dis,
                           const float* __restrict__ bias, float* __restrict__ acc, int n) {
    int i = blockIdx.x * blockDim.x + threadIdx.x;
    if (i >= n) return;
    float d2 = dis[i] * dis[i];
#pragma unroll
    for (int f = 0; f < OUTF; ++f)
        acc[i * OUTF + f] = bias[f] + ht[i * OUTF + f] * d2;
}

// ---------------- per-edge gather * norm -> scatter-add (L2-resident atomics) ----
template <int OUTF>
__global__ void k_scatter(const long long* __restrict__ row, const long long* __restrict__ col,
                          const float* __restrict__ dis, const float* __restrict__ ht,
                          float* acc, int e) {
    int i = blockIdx.x * blockDim.x + threadIdx.x;
    if (i >= e) return;
    long long r = row[i], c = col[i];
    float w = dis[r] * dis[c];
#pragma unroll
    for (int f = 0; f < OUTF; ++f)
        atomicAdd(&acc[c * OUTF + f], ht[r * OUTF + f] * w);
}

// ---------------- MLP head: [256,4096]@[4096,32] via K-chained WMMA ----------------
__global__ void k_mlp1(const float* __restrict__ X, const float* __restrict__ W,
                       const float* __restrict__ bias, float* __restrict__ out) {
    int wave  = (blockIdx.x * blockDim.x + threadIdx.x) >> 5;   // 0..31
    int lane  = threadIdx.x & 31;
    int mtile = wave >> 1;
    int ntile = wave & 1;
    int mrow  = lane & 15;
    int hi    = lane >> 4;
    int khalf = hi * 2;
    int ncol  = ntile * 16 + (lane & 15);

    v8f c = {};
    const float* xr = X + (mtile * 16 + mrow) * 4096;
    for (int k0 = 0; k0 < 4096; k0 += 4) {
        int kA = k0 + khalf;
        v2f a, b;
        a.x = xr[kA + 0];
        a.y = xr[kA + 1];
        b.x = W[(kA + 0) * 32 + ncol];
        b.y = W[(kA + 1) * 32 + ncol];
        c = __builtin_amdgcn_wmma_f32_16x16x4_f32(false, a, false, b, (short)0, c,
                                                  false, false);
    }
#pragma unroll
    for (int r = 0; r < 8; ++r) {
        int m = mtile * 16 + r + hi * 8;
        out[m * 32 + ncol] = lrelu(c[r] + bias[ncol]);
    }
}

__global__ void k_mlp23(const float* __restrict__ h1, const float* __restrict__ W2,
                        const float* __restrict__ b2, const float* __restrict__ W3,
                        const float* __restrict__ b3, float* __restrict__ out) {
    int r = blockIdx.x * blockDim.x + threadIdx.x;
    if (r >= BATCH) return;
    float h2[16];
#pragma unroll
    for (int j = 0; j < 16; ++j) {
        float s = b2[j];
        for (int k = 0; k < 32; ++k) s += h1[r * 32 + k] * W2[k * 16 + j];
        h2[j] = lrelu(s);
    }
#pragma unroll
    for (int j = 0; j < 3; ++j) {
        float s = b3[j];
        for (int k = 0; k < 16; ++k) s += h2[k] * W3[k * 3 + j];
        out[r * 3 + j] = lrelu(s);
    }
}

extern "C" void kernel_launch(void* const* d_in, const int* in_sizes, int n_in,
                              void* d_out, int out_size, void* d_ws, size_t ws_size,
                              hipStream_t stream) {
    const float*     z     = (const float*)d_in[0];
    const long long* ei    = (const long long*)d_in[1];
    const float*     W_inv = (const float*)d_in[2];
    const float*     b_inv = (const float*)d_in[3];
    const float*     W1    = (const float*)d_in[4];
    const float*     b1    = (const float*)d_in[5];
    const float*     W2    = (const float*)d_in[6];
    const float*     b2    = (const float*)d_in[7];
    const float*     W3    = (const float*)d_in[8];
    const float*     b3    = (const float*)d_in[9];
    const float*     Wl1   = (const float*)d_in[10];
    const float*     bl1   = (const float*)d_in[11];
    const float*     Wl2   = (const float*)d_in[12];
    const float*     bl2   = (const float*)d_in[13];
    const float*     Wl3   = (const float*)d_in[14];
    const float*     bl3   = (const float*)d_in[15];
    float*           outp  = (float*)d_out;

    const int E = in_sizes[1] / 2;
    const long long* row = ei;
    const long long* col = ei + E;

    // workspace layout (floats)
    float* ws   = (float*)d_ws;
    float* dis  = ws;                 // N
    float* bufX = dis + NN;           // 16N
    float* bufA = bufX + 16 * NN;     // 8N
    float* bufB = bufA + 8 * NN;      // 8N
    float* out1 = bufB + 8 * NN;      // 256*32

    const int TPB = 256;
    const int nBlk = NN / TPB;
    const int eBlk = (E + TPB - 1) / TPB;
    const int tiles = NN / 16;        // 16384
    const int tBlk = tiles / 4;       // 4 waves (128 thr) per block

    // deg^{-1/2}
    k_fill1<<<nBlk, TPB, 0, stream>>>(dis, NN);
    k_deg<<<eBlk, TPB, 0, stream>>>(col, dis, E);
    k_rsqrt<<<nBlk, TPB, 0, stream>>>(dis, NN);

    // h0 = lrelu(z @ W_inv + b_inv)  [N,16]
    k_transform<3, 16, true><<<tBlk, 128, 0, stream>>>(z, W_inv, b_inv, bufX, tiles);

    // conv1: 16 -> 4
    k_transform<16, 4, false><<<tBlk, 128, 0, stream>>>(bufX, W1, nullptr, bufA, tiles);
    k_acc_init<4><<<nBlk, TPB, 0, stream>>>(bufA, dis, b1, bufB, NN);
    k_scatter<4><<<eBlk, TPB, 0, stream>>>(row, col, dis, bufA, bufB, E);

    // conv2: 4 -> 8
    k_transform<4, 8, false><<<tBlk, 128, 0, stream>>>(bufB, W2, nullptr, bufA, tiles);
    k_acc_init<8><<<nBlk, TPB, 0, stream>>>(bufA, dis, b2, bufX, NN);
    k_scatter<8><<<eBlk, TPB, 0, stream>>>(row, col, dis, bufA, bufX, E);

    // conv3: 8 -> 4
    k_transform<8, 4, false><<<tBlk, 128, 0, stream>>>(bufX, W3, nullptr, bufA, tiles);
    k_acc_init<4><<<nBlk, TPB, 0, stream>>>(bufA, dis, b3, bufB, NN);
    k_scatter<4><<<eBlk, TPB, 0, stream>>>(row, col, dis, bufA, bufB, E);

    // MLP head
    k_mlp1<<<8, 128, 0, stream>>>(bufB, Wl1, bl1, out1);       // [256,4096]@[4096,32]
    k_mlp23<<<1, 256, 0, stream>>>(out1, Wl2, bl2, Wl3, bl3, outp);

    // second tuple output: edge_index passthrough (raw bytes after the 768 floats)
    hipMemcpyAsync((char*)d_out + (size_t)BATCH * 3 * sizeof(float), d_in[1],
                   (size_t)2 * E * sizeof(long long), hipMemcpyDeviceToDevice, stream);
}